// LongTextEncoder_19284403159429
// MI455X (gfx1250) — compile-verified
//
#include <hip/hip_runtime.h>
#include <hip/hip_bf16.h>
#include <math.h>

#define N_SAMP 512
#define SEQ_L  512
#define HDIM   768
#define NTEXT  32

typedef float v2f __attribute__((ext_vector_type(2)));
typedef float v8f __attribute__((ext_vector_type(8)));

// ---------------------------------------------------------------------------
// Kernel 1: mean over sequence dim.  hidden_states [N,L,H] -> mean [N,H]
// One block per n; 192 threads (6 waves), one float4 per thread per row:
// 192 * 4 = 768 floats = one full row per iteration, perfectly coalesced.
// ---------------------------------------------------------------------------
__global__ void mean_kernel(const float* __restrict__ hs, float* __restrict__ mean)
{
    const int n   = blockIdx.x;
    const int tid = threadIdx.x;            // 192 threads
    const float4* base4 = (const float4*)(hs + (size_t)n * SEQ_L * HDIM);
    float4 acc = make_float4(0.f, 0.f, 0.f, 0.f);
    for (int l = 0; l < SEQ_L; ++l) {
        const float4 v = base4[(size_t)l * (HDIM / 4) + tid];
        acc.x += v.x; acc.y += v.y; acc.z += v.z; acc.w += v.w;
    }
    const float inv = 1.0f / (float)SEQ_L;
    acc.x *= inv; acc.y *= inv; acc.z *= inv; acc.w *= inv;
    ((float4*)mean)[(size_t)n * (HDIM / 4) + tid] = acc;
}

// ---------------------------------------------------------------------------
// Kernel 2/3: fp32 GEMM via V_WMMA_F32_16X16X4_F32.
//   D[M,Nc] = A[M,K] * B (+ bias per column)
//   TRANS_B=1: B given as Bt[Nc,K], i.e. B[k,j] = Bt[j,k]   (for q = mean@Wq^T)
//   TRANS_B=0: B given row-major [K,Nc]                      (for qk = q@Wk)
// One wave computes one 16x16 tile; K-loop in steps of 4.
// A layout (ISA 7.12.2, 32-bit A 16x4): lanes 0-15: VGPR0=K0,VGPR1=K1;
//                                       lanes16-31: VGPR0=K2,VGPR1=K3.
// B layout symmetric; C/D: VGPR r -> M=r (lanes 0-15), M=r+8 (lanes 16-31).
// ---------------------------------------------------------------------------
template <bool TRANS_B, bool ADD_BIAS>
__global__ void gemm_wmma_f32(const float* __restrict__ A,
                              const float* __restrict__ B,
                              const float* __restrict__ bias,
                              float* __restrict__ D,
                              int M, int Nc, int K)
{
    const int lane   = threadIdx.x & 31;
    const int wave   = threadIdx.x >> 5;
    const int waveId = blockIdx.x * (blockDim.x >> 5) + wave;
    const int tilesN = Nc >> 4;
    const int tilesM = M  >> 4;
    if (waveId >= tilesM * tilesN) return;   // wave-uniform: EXEC stays all-1s for WMMA

    const int tm = waveId / tilesN;
    const int tn = waveId % tilesN;
    const int m0 = tm << 4, n0 = tn << 4;
    const int half = lane >> 4;              // 0: lanes 0-15, 1: lanes 16-31
    const int r16  = lane & 15;

    v8f acc = {};
    const float* Arow = A + (size_t)(m0 + r16) * K + 2 * half;
    for (int k0 = 0; k0 < K; k0 += 4) {
        v2f a;
        a.x = Arow[k0];
        a.y = Arow[k0 + 1];
        v2f b;
        if (TRANS_B) {
            const float* Bp = B + (size_t)(n0 + r16) * K + k0 + 2 * half;
            b.x = Bp[0];
            b.y = Bp[1];
        } else {
            const float* Bp = B + (size_t)(k0 + 2 * half) * Nc + n0 + r16;
            b.x = Bp[0];
            b.y = Bp[Nc];
        }
        acc = __builtin_amdgcn_wmma_f32_16x16x4_f32(
            /*neg_a=*/false, a, /*neg_b=*/false, b,
            /*c_mod=*/(short)0, acc, /*reuse_a=*/false, /*reuse_b=*/false);
    }

    const float bv = ADD_BIAS ? bias[n0 + r16] : 0.0f;
#pragma unroll
    for (int r = 0; r < 8; ++r) {
        const int row = m0 + r + 8 * half;
        D[(size_t)row * Nc + n0 + r16] = acc[r] + bv;
    }
}

// ---------------------------------------------------------------------------
// Kernel 4: per-n fused  scores -> masked softmax -> attention pooling.
// One block per n (256 threads = 8 waves).
//   phase 1: wave w handles l in [w*64, w*64+64); lanes dot 768-wide rows
//            against qk[n] (held in registers as 6 float4 per lane).
//   phase 2: block softmax over 512 scores in LDS.
//   phase 3: pooled[n,h] = sum_l alpha[l]*h[n,l,h]  (row re-read is L2-hot;
//            threads 0..191 each own one float4 column group).
// ---------------------------------------------------------------------------
__global__ void attn_pool_kernel(const float* __restrict__ hs,
                                 const int* __restrict__ mask,
                                 const float* __restrict__ qk,
                                 float* __restrict__ pooled)
{
    __shared__ float qk_sh[HDIM];
    __shared__ float sc[SEQ_L];
    __shared__ float red[256];

    const int n    = blockIdx.x;
    const int tid  = threadIdx.x;
    const int lane = tid & 31;
    const int wave = tid >> 5;
    const size_t base = (size_t)n * SEQ_L * HDIM;

    if (tid < HDIM / 4)
        ((float4*)qk_sh)[tid] = ((const float4*)qk)[(size_t)n * (HDIM / 4) + tid];
    __syncthreads();

    // ---- phase 1: scores ----
    const float4* qk4 = (const float4*)qk_sh;
    float4 qv[6];
#pragma unroll
    for (int c = 0; c < 6; ++c) qv[c] = qk4[c * 32 + lane];

    const float rscale = 0.03608439182435161f;   // 1/sqrt(768)
    for (int l = wave * 64; l < wave * 64 + 64; ++l) {
        const float4* row4 = (const float4*)(hs + base + (size_t)l * HDIM);
        float acc = 0.f;
#pragma unroll
        for (int c = 0; c < 6; ++c) {
            const float4 hv = row4[c * 32 + lane];
            acc += hv.x * qv[c].x + hv.y * qv[c].y + hv.z * qv[c].z + hv.w * qv[c].w;
        }
#pragma unroll
        for (int off = 16; off > 0; off >>= 1) acc += __shfl_xor(acc, off, 32);
        if (lane == 0)
            sc[l] = (mask[(size_t)n * SEQ_L + l] == 0) ? -1e9f : acc * rscale;
    }
    __syncthreads();

    // ---- phase 2: masked softmax over sc[0..511] ----
    red[tid] = fmaxf(sc[tid], sc[tid + 256]);
    __syncthreads();
    for (int s = 128; s > 0; s >>= 1) {
        if (tid < s) red[tid] = fmaxf(red[tid], red[tid + s]);
        __syncthreads();
    }
    const float mx = red[0];
    __syncthreads();
    const float e0 = expf(sc[tid] - mx);
    const float e1 = expf(sc[tid + 256] - mx);
    red[tid] = e0 + e1;
    __syncthreads();
    for (int s = 128; s > 0; s >>= 1) {
        if (tid < s) red[tid] += red[tid + s];
        __syncthreads();
    }
    const float inv = 1.0f / red[0];
    __syncthreads();
    sc[tid]       = e0 * inv;
    sc[tid + 256] = e1 * inv;
    __syncthreads();

    // ---- phase 3: weighted pooling (float4 per thread, 192 active) ----
    if (tid < HDIM / 4) {
        const float4* base4 = (const float4*)(hs + base);
        float4 acc = make_float4(0.f, 0.f, 0.f, 0.f);
        for (int l = 0; l < SEQ_L; ++l) {
            const float al = sc[l];
            const float4 v = base4[(size_t)l * (HDIM / 4) + tid];
            acc.x += al * v.x; acc.y += al * v.y;
            acc.z += al * v.z; acc.w += al * v.w;
        }
        ((float4*)pooled)[(size_t)n * (HDIM / 4) + tid] = acc;
    }
}

// ---------------------------------------------------------------------------
// Kernel 5: segment mean.  sample_map is sorted, so each text t owns a
// contiguous [lo,hi) range of n — no atomics, fully deterministic.
// ---------------------------------------------------------------------------
__global__ void seg_mean_kernel(const float* __restrict__ pooled,
                                const long long* __restrict__ smap,
                                float* __restrict__ out)
{
    __shared__ int s_lo, s_hi;
    const int t = blockIdx.x;
    if (threadIdx.x == 0) {
        int lo = 0;
        while (lo < N_SAMP && smap[lo] < (long long)t) ++lo;
        int hi = lo;
        while (hi < N_SAMP && smap[hi] == (long long)t) ++hi;
        s_lo = lo; s_hi = hi;
    }
    __syncthreads();
    const int lo = s_lo, hi = s_hi;
    const int cnt = hi - lo;
    const float inv = 1.0f / (float)(cnt < 1 ? 1 : cnt);
    for (int h = threadIdx.x; h < HDIM; h += blockDim.x) {
        float acc = 0.f;
        for (int n = lo; n < hi; ++n) acc += pooled[(size_t)n * HDIM + h];
        out[(size_t)t * HDIM + h] = acc * inv;
    }
}

// ---------------------------------------------------------------------------
extern "C" void kernel_launch(void* const* d_in, const int* in_sizes, int n_in,
                              void* d_out, int out_size, void* d_ws, size_t ws_size,
                              hipStream_t stream)
{
    (void)in_sizes; (void)n_in; (void)out_size; (void)ws_size;

    const float*     hs   = (const float*)d_in[0];
    const int*       mask = (const int*)d_in[1];
    const long long* smap = (const long long*)d_in[2];
    const float*     Wq   = (const float*)d_in[3];
    const float*     bq   = (const float*)d_in[4];
    const float*     Wk   = (const float*)d_in[5];
    // d_in[6] = bk: adds a per-n constant to all scores -> softmax-invariant, omitted.
    float* out = (float*)d_out;

    float* mean_qk = (float*)d_ws;                 // [512,768]; reused for qk
    float* qbuf    = mean_qk + N_SAMP * HDIM;      // [512,768]
    float* pooled  = qbuf    + N_SAMP * HDIM;      // [512,768]

    // 1) mean over L (192 threads = 32 float4 lanes per wave * 6 waves)
    mean_kernel<<<N_SAMP, 192, 0, stream>>>(hs, mean_qk);

    // 2) q = mean @ Wq^T + bq     (tiles: 32x48 = 1536 waves = 192 blocks x 8 waves)
    gemm_wmma_f32<true, true><<<192, 256, 0, stream>>>(mean_qk, Wq, bq, qbuf,
                                                       N_SAMP, HDIM, HDIM);

    // 3) qk = q @ Wk   (written over the dead mean buffer)
    gemm_wmma_f32<false, false><<<192, 256, 0, stream>>>(qbuf, Wk, nullptr, mean_qk,
                                                         N_SAMP, HDIM, HDIM);

    // 4) fused scores/softmax/pool per n
    attn_pool_kernel<<<N_SAMP, 256, 0, stream>>>(hs, mask, mean_qk, pooled);

    // 5) deterministic segment mean (sample_map sorted)
    seg_mean_kernel<<<NTEXT, 256, 0, stream>>>(pooled, smap, out);
}